// DWT_2D_38465727103451
// MI455X (gfx1250) — compile-verified
//
#include <hip/hip_runtime.h>

// 2D Haar DWT, level 1:  x:(8,96,512,512) f32  ->  out:(8,384,256,256) f32
// Bands on channel axis in order [ll, lh, hl, hh].
// Memory-bound: ~1.61 GB traffic -> ~69us floor at 23.3 TB/s.
// Data path: async global->LDS staging (ASYNCcnt) + non-temporal wide stores.

typedef float v4f __attribute__((ext_vector_type(4)));

#define IN_W      512
#define OUT_W     256
#define PLANE_IN  (512 * 512)
#define PLANE_OUT (256 * 256)
#define C_DIM     96

__global__ __launch_bounds__(256) void dwt_haar_kernel(
    const float* __restrict__ x, float* __restrict__ out) {
  // 8 input rows x 512 cols of f32 = 16 KB staged per block
  __shared__ __align__(16) float tile[8 * IN_W];

  const unsigned t  = threadIdx.x;       // 0..255
  const unsigned b  = blockIdx.x;        // 0..49151
  const unsigned ib = b & 63u;           // output-row group within plane (4 rows each)
  const unsigned nc = b >> 6;            // n*96 + c, 0..767
  const unsigned n  = nc / C_DIM;        // batch index

  // ---- Stage 8 input rows into LDS with the gfx1250 async engine ----
  // Each thread issues 4x b128 async loads; per-wave each instr moves a
  // contiguous 512B global run into a contiguous 512B LDS run.
  const size_t inBase = (size_t)nc * PLANE_IN + (size_t)(ib * 8u) * IN_W;
#pragma unroll
  for (int q = 0; q < 4; ++q) {
    const unsigned v    = (unsigned)q * 256u + t;  // float4 index 0..1023
    const unsigned row  = v >> 7;                  // 0..7  (128 float4 per row)
    const unsigned col4 = v & 127u;
    const float* gp = x + inBase + (size_t)row * IN_W + (size_t)col4 * 4u;
    const unsigned lds = (unsigned)(size_t)(&tile[v * 4u]);  // LDS byte offset
    asm volatile("global_load_async_to_lds_b128 %0, %1, off th:TH_LOAD_NT"
                 :: "v"(lds), "v"(gp)
                 : "memory");
  }
  asm volatile("s_wait_asynccnt 0" ::: "memory");
  __syncthreads();

  // ---- Butterfly: each thread produces 4 output columns for one row ----
  const unsigned di = t >> 6;      // 0..3 : output row within block
  const unsigned tc = t & 63u;     // 0..63: column group
  const unsigned j0 = tc * 4u;     // first output column

  const float* er = &tile[di * (2u * IN_W) + tc * 8u];  // even input row 2*di
  const float* od = er + IN_W;                          // odd  input row 2*di+1
  const v4f r0 = *(const v4f*)(er);
  const v4f r1 = *(const v4f*)(er + 4);
  const v4f r2 = *(const v4f*)(od);
  const v4f r3 = *(const v4f*)(od + 4);

  // a = even-row even cols, b = even-row odd cols, c = odd-row even, d = odd-row odd
  const v4f a = __builtin_shufflevector(r0, r1, 0, 2, 4, 6);
  const v4f bb = __builtin_shufflevector(r0, r1, 1, 3, 5, 7);
  const v4f c = __builtin_shufflevector(r2, r3, 0, 2, 4, 6);
  const v4f d = __builtin_shufflevector(r2, r3, 1, 3, 5, 7);

  const v4f s0 = a + bb, d0 = a - bb;   // row butterflies
  const v4f s1 = c + d,  d1 = c - d;
  const float q4 = 0.25f;
  const v4f ll = q4 * (s0 + s1);
  const v4f lh = q4 * (s0 - s1);
  const v4f hl = q4 * (d0 + d1);
  const v4f hh = q4 * (d0 - d1);

  // ---- Non-temporal coalesced stores, one float4 per band ----
  // output plane index for band k: n*384 + k*96 + c  ==  nc + 288*n + 96*k
  const unsigned i = ib * 4u + di;
  const size_t rowOff = (size_t)i * OUT_W + j0;
  float* o = out + ((size_t)nc + (size_t)n * 288u) * PLANE_OUT + rowOff;
  const size_t bandStride = (size_t)C_DIM * PLANE_OUT;

  __builtin_nontemporal_store(ll, (v4f*)(o + 0 * bandStride));
  __builtin_nontemporal_store(lh, (v4f*)(o + 1 * bandStride));
  __builtin_nontemporal_store(hl, (v4f*)(o + 2 * bandStride));
  __builtin_nontemporal_store(hh, (v4f*)(o + 3 * bandStride));
}

extern "C" void kernel_launch(void* const* d_in, const int* in_sizes, int n_in,
                              void* d_out, int out_size, void* d_ws, size_t ws_size,
                              hipStream_t stream) {
  (void)in_sizes; (void)n_in; (void)out_size; (void)d_ws; (void)ws_size;
  const float* x = (const float*)d_in[0];
  float* out = (float*)d_out;
  // blocks: (8*96 planes) * (256 out rows / 4 rows per block) = 768 * 64 = 49152
  dim3 grid(8 * 96 * 64);
  dim3 block(256);
  dwt_haar_kernel<<<grid, block, 0, stream>>>(x, out);
}